// SmoothLayer_5669356831096
// MI455X (gfx1250) — compile-verified
//
#include <hip/hip_runtime.h>
#include <hip/hip_bf16.h>
#include <stdint.h>

// ---------------------------------------------------------------------------
// 7-point 3D stencil (zero-padded Laplacian) * per-voxel weights + residual.
//   out[b,c,z,y,x] = x + w[c,z,y,x] * (-6*x + 6 neighbors)
// B=2, C=48, D=96, fp32.  Memory-bound: ~1.0 GB total traffic -> ~45us
// roofline at 23.3 TB/s.  CDNA5 path used: async global->LDS staging
// (GLOBAL_LOAD_ASYNC_TO_LDS_B32 + s_wait_asynccnt) with a 4-plane LDS ring.
// ---------------------------------------------------------------------------

#define DD        96
#define CC        48
#define NVOL      (2 * CC)          // B*C independent volumes
#define TX        32
#define TY        8
#define NTHREADS  (TX * TY)         // 256 threads = 8 wave32 waves
#define HALO_W    (TX + 2)          // 34
#define HALO_H    (TY + 2)          // 10
#define PLANE     (HALO_W * HALO_H) // 340 floats per halo plane
#define NSLOTS    4                 // ring depth (z-1, z, z+1 live + z+2 inflight)

// Async copy one fp32 from global memory into LDS at a 32-bit LDS byte offset.
// GV addressing mode: 64-bit VGPR address, SADDR = off.  Tracked by ASYNCcnt.
__device__ __forceinline__ void async_load_f32_to_lds(uint32_t lds_byte_off,
                                                      const float* g) {
  asm volatile("global_load_async_to_lds_b32 %0, %1, off"
               :
               : "v"(lds_byte_off), "v"((uint64_t)(uintptr_t)g)
               : "memory");
}

__device__ __forceinline__ void wait_async_done() {
#if __has_builtin(__builtin_amdgcn_s_wait_asynccnt)
  __builtin_amdgcn_s_wait_asynccnt(0);
#else
  asm volatile("s_wait_asynccnt 0x0" ::: "memory");
#endif
}

__global__ __launch_bounds__(NTHREADS) void smooth3d_kernel(
    const float* __restrict__ x, const float* __restrict__ w,
    float* __restrict__ out) {
  __shared__ float smem[NSLOTS * PLANE];

  const int tid = (int)threadIdx.x;
  const int x0  = (int)blockIdx.x * TX;
  const int y0  = (int)blockIdx.y * TY;
  const int vol = (int)blockIdx.z;      // b*C + c
  const int c   = vol % CC;

  const float* __restrict__ xvol = x + (size_t)vol * (DD * DD * DD);
  const float* __restrict__ wvol = w + (size_t)c   * (DD * DD * DD);
  float* __restrict__ ovol       = out + (size_t)vol * (DD * DD * DD);

  // Per-thread halo-plane load assignments (z-invariant): elements tid and
  // tid+256 of the 340-element 34x10 halo plane.
  const int  e0   = tid;
  const int  ey0  = e0 / HALO_W, ex0 = e0 % HALO_W;
  const int  gx0  = x0 - 1 + ex0, gy0 = y0 - 1 + ey0;
  const bool in0  = ((unsigned)gx0 < (unsigned)DD) && ((unsigned)gy0 < (unsigned)DD);
  const int  goff0 = gy0 * DD + gx0;

  const int  e1   = tid + NTHREADS;
  const bool has1 = (e1 < PLANE);
  const int  ey1  = e1 / HALO_W, ex1 = e1 % HALO_W;
  const int  gx1  = x0 - 1 + ex1, gy1 = y0 - 1 + ey1;
  const bool in1  = has1 && ((unsigned)gx1 < (unsigned)DD) && ((unsigned)gy1 < (unsigned)DD);
  const int  goff1 = gy1 * DD + gx1;

  // Prologue: clear the whole ring.  This provides (a) the zero plane for
  // z = -1 and (b) permanently-zero x/y-OOB halo cells (their OOB-ness is
  // z-invariant, so async loads never overwrite them).
  for (int i = tid; i < NSLOTS * PLANE; i += NTHREADS) smem[i] = 0.0f;
  __syncthreads();

  // Plane p lives in ring slot (p+1) & 3.
  auto issue_plane = [&](int p) {
    const int slot = ((p + 1) & (NSLOTS - 1)) * PLANE;
    if (p < DD) {
      const float* gp = xvol + (size_t)p * (DD * DD);
      if (in0) async_load_f32_to_lds((uint32_t)(size_t)&smem[slot + e0], gp + goff0);
      if (in1) async_load_f32_to_lds((uint32_t)(size_t)&smem[slot + e1], gp + goff1);
    } else {
      // z out of range: whole plane is zero padding.
      smem[slot + e0] = 0.0f;
      if (has1) smem[slot + e1] = 0.0f;
    }
  };

  issue_plane(0);
  issue_plane(1);

  const int lx  = tid & (TX - 1);
  const int ly  = tid / TX;
  const int ctr = (ly + 1) * HALO_W + (lx + 1);
  const int gxy = (y0 + ly) * DD + (x0 + lx);

  for (int z = 0; z < DD; ++z) {
    // Plane z+1's async loads (issued last iteration / prologue) must land,
    // and all waves must be past reading the slot we are about to recycle.
    wait_async_done();
    __syncthreads();

    // Start streaming plane z+2 into the slot that held plane z-2.
    issue_plane(z + 2);

    const int sm = ((z)     & (NSLOTS - 1)) * PLANE;  // plane z-1
    const int sc = ((z + 1) & (NSLOTS - 1)) * PLANE;  // plane z
    const int sp = ((z + 2) & (NSLOTS - 1)) * PLANE;  // plane z+1

    const float cen = smem[sc + ctr];
    const float xm  = smem[sc + ctr - 1];
    const float xp  = smem[sc + ctr + 1];
    const float ym  = smem[sc + ctr - HALO_W];
    const float yp  = smem[sc + ctr + HALO_W];
    const float zm  = smem[sm + ctr];
    const float zp  = smem[sp + ctr];

    const float wv  = wvol[(size_t)z * (DD * DD) + gxy];
    const float lap = fmaf(-6.0f, cen, (xm + xp) + ((ym + yp) + (zm + zp)));
    const float res = fmaf(wv, lap, cen);

    __builtin_nontemporal_store(res, &ovol[(size_t)z * (DD * DD) + gxy]);
  }
}

extern "C" void kernel_launch(void* const* d_in, const int* in_sizes, int n_in,
                              void* d_out, int out_size, void* d_ws, size_t ws_size,
                              hipStream_t stream) {
  (void)in_sizes; (void)n_in; (void)out_size; (void)d_ws; (void)ws_size;
  const float* x = (const float*)d_in[0];
  const float* w = (const float*)d_in[1];
  float* out     = (float*)d_out;

  dim3 grid(DD / TX, DD / TY, NVOL);   // (3, 12, 96) = 3456 blocks
  smooth3d_kernel<<<grid, NTHREADS, 0, stream>>>(x, w, out);
}